// NetModel_53755810676778
// MI455X (gfx1250) — compile-verified
//
#include <hip/hip_runtime.h>
#include <hip/hip_bf16.h>
#include <math.h>

// ---- problem constants (match reference) ----
#define N_AP      10000
#define N_STA     40000
#define N_NODE    50000
#define E_AA      200000
#define E_AS      500000
#define E_SA      500000
#define E_TOT     1200000
#define DIM       128
#define D_NODE_IN 21
#define K_NODE_L0 24      // D_NODE_IN padded to a multiple of 4 (zero pad)
#define D_EDGE_IN 4
#define NEG_SLOPE 0.01f
#define EPS_BN    1e-5f

typedef __attribute__((ext_vector_type(2))) float v2f;
typedef __attribute__((ext_vector_type(8))) float v8f;

#if __has_builtin(__builtin_amdgcn_wmma_f32_16x16x4_f32)
#define HAVE_WMMA_F32 1
#endif

// ------------------------------------------------------------------ fills
__global__ void k_fill_f32(float* p, float v, long long n) {
  long long i = (long long)blockIdx.x * blockDim.x + threadIdx.x;
  if (i < n) p[i] = v;
}
__global__ void k_fill_u32(unsigned* p, unsigned v, long long n) {
  long long i = (long long)blockIdx.x * blockDim.x + threadIdx.x;
  if (i < n) p[i] = v;
}

// ------------------------------------------------------------------ batchnorm
__global__ void k_bn_stats(const float* __restrict__ x, int N, int D,
                           float* __restrict__ acc) {
  int t = threadIdx.x;            // blockDim.x == 128 >= D
  if (t >= D) return;
  float s = 0.f, s2 = 0.f;
  for (int r = blockIdx.x; r < N; r += gridDim.x) {
    float v = x[(size_t)r * D + t];
    s += v; s2 += v * v;
  }
  atomicAdd(&acc[t], s);
  atomicAdd(&acc[D + t], s2);
}

// reads x[rows x D], writes y with row stride Dout (Dout >= D; pad untouched)
__global__ void k_bn_apply(const float* __restrict__ x, long long total, int D,
                           int Dout, const float* __restrict__ acc, float ninv,
                           float* __restrict__ y) {
  long long i = (long long)blockIdx.x * blockDim.x + threadIdx.x;
  if (i >= total) return;
  long long row = i / D;
  int c = (int)(i - row * D);
  float m  = acc[c] * ninv;
  float va = acc[D + c] * ninv - m * m;
  float rs = rsqrtf(va + EPS_BN);
  y[row * Dout + c] = (x[i] - m) * rs;
}

// ------------------------------------------------------------------ WMMA GEMM
// W[K][128] -> Wt[128][K] so B fragments are contiguous b64 loads
__global__ void k_transpose(const float* __restrict__ W, float* __restrict__ Wt,
                            int K) {
  int i = blockIdx.x * blockDim.x + threadIdx.x;
  if (i >= K * DIM) return;
  int k = i / DIM, c = i - k * DIM;
  Wt[(size_t)c * K + k] = W[i];
}

// Y[M x 128] = X[M x K] @ W + bias.  One wave computes 32 rows x 128 cols with
// V_WMMA_F32_16X16X4_F32; each B fragment (b64) feeds two WMMAs.
// f32 A 16x4 layout: lanes 0-15 M=0..15 hold {K=k,k+1}, lanes 16-31 {k+2,k+3}.
template <int K>
__global__ void __launch_bounds__(256)
k_gemm_wmma(const float* __restrict__ X, const float* __restrict__ Wt,
            const float* __restrict__ bias, float* __restrict__ Y, int M) {
  int wave  = blockIdx.x * (blockDim.x >> 5) + (threadIdx.x >> 5);
  int lane  = threadIdx.x & 31;
  int mBase = wave * 32;
  if (mBase >= M) return;
  int half = lane >> 4;
  int l16  = lane & 15;
  int m0 = mBase + l16;
  int m1 = mBase + 16 + l16;
  m0 = (m0 < M) ? m0 : (M - 1);      // clamp reads; stores handle the tail
  m1 = (m1 < M) ? m1 : (M - 1);
  const float* x0 = X + (size_t)m0 * K + half * 2;
  const float* x1 = X + (size_t)m1 * K + half * 2;

  v8f acc[16];
  const v8f vz = {0.f,0.f,0.f,0.f,0.f,0.f,0.f,0.f};
#pragma unroll
  for (int i = 0; i < 16; ++i) acc[i] = vz;

  for (int k0 = 0; k0 < K; k0 += 4) {
    v2f a0 = *(const v2f*)(x0 + k0);
    v2f a1 = *(const v2f*)(x1 + k0);
#pragma unroll
    for (int nt = 0; nt < 8; ++nt) {
      int col = nt * 16 + l16;
      v2f b = *(const v2f*)(Wt + (size_t)col * K + k0 + half * 2);
#ifdef HAVE_WMMA_F32
      acc[nt] = __builtin_amdgcn_wmma_f32_16x16x4_f32(
          false, a0, false, b, (short)0, acc[nt], false, false);
      acc[8 + nt] = __builtin_amdgcn_wmma_f32_16x16x4_f32(
          false, a1, false, b, (short)0, acc[8 + nt], false, false);
#else
      acc[nt][0]     += a0.x * b.x + a0.y * b.y;  // compile-safe placeholder
      acc[8 + nt][0] += a1.x * b.x + a1.y * b.y;
#endif
    }
  }

  // D layout: vgpr r -> M = r + 8*half, N = l16 (per 16x16 tile)
  if (mBase + 32 <= M) {
    // uniform fast path: all rows in range -> unconditional stores
#pragma unroll
    for (int nt = 0; nt < 8; ++nt) {
      int col = nt * 16 + l16;
      float bv = bias ? bias[col] : 0.f;
      float* y0 = Y + (size_t)(mBase + half * 8) * DIM + col;
#pragma unroll
      for (int r = 0; r < 8; ++r) {
        y0[(size_t)r * DIM]        = acc[nt][r] + bv;
        y0[(size_t)(r + 16) * DIM] = acc[8 + nt][r] + bv;
      }
    }
  } else {
    // tail wave only
#pragma unroll
    for (int nt = 0; nt < 8; ++nt) {
      int col = nt * 16 + l16;
      float bv = bias ? bias[col] : 0.f;
#pragma unroll
      for (int r = 0; r < 8; ++r) {
        int row0 = mBase + half * 8 + r;
        int row1 = row0 + 16;
        if (row0 < M) Y[(size_t)row0 * DIM + col] = acc[nt][r] + bv;
        if (row1 < M) Y[(size_t)row1 * DIM + col] = acc[8 + nt][r] + bv;
      }
    }
  }
}

// ------------------------------------------------------------------ edge passes
__device__ __forceinline__ unsigned f32_key(float f) {
  unsigned u = __float_as_uint(f);
  return (u & 0x80000000u) ? ~u : (u | 0x80000000u);
}
__device__ __forceinline__ float key_f32(unsigned k) {
  unsigned u = (k & 0x80000000u) ? (k & 0x7FFFFFFFu) : ~k;
  return __uint_as_float(u);
}
__device__ __forceinline__ float lrelu(float v) {
  return (v > 0.f) ? v : NEG_SLOPE * v;
}

// pass A: f_out = lrelu(F[src]+G[dst]+(efeat@W_fij+bias_e)); score; seg-max
__global__ void __launch_bounds__(256)
k_edge_attn(const int* __restrict__ src, const int* __restrict__ dst, int E,
            int srcOff, int dstOff,
            const float* __restrict__ F, const float* __restrict__ G,
            float* __restrict__ EW, const float* __restrict__ attn,
            float* __restrict__ score, unsigned* __restrict__ segmax) {
  int e = blockIdx.x * 8 + (threadIdx.x >> 5);
  if (e >= E) return;
  int lane = threadIdx.x & 31;
  int sr = src[e] + srcOff;
  int dr = dst[e] + dstOff;
  float4 f = ((const float4*)(F + (size_t)sr * DIM))[lane];
  float4 g = ((const float4*)(G + (size_t)dr * DIM))[lane];
  float4* er = (float4*)(EW + (size_t)e * DIM) + lane;
  float4 w = *er;
  float4 a4 = ((const float4*)attn)[lane];
  float4 v;
  v.x = lrelu(f.x + g.x + w.x);
  v.y = lrelu(f.y + g.y + w.y);
  v.z = lrelu(f.z + g.z + w.z);
  v.w = lrelu(f.w + g.w + w.w);
  *er = v;
  float s = v.x * a4.x + v.y * a4.y + v.z * a4.z + v.w * a4.w;
#pragma unroll
  for (int o = 16; o > 0; o >>= 1) s += __shfl_xor(s, o, 32);
  if (lane == 0) {
    score[e] = s;
    atomicMax(&segmax[dr], f32_key(s));
  }
}

// pass B: ex = exp(s - max[dst]); seg-sum
__global__ void k_edge_expsum(const int* __restrict__ dst, int E, int dstOff,
                              float* __restrict__ score,
                              const unsigned* __restrict__ segmax,
                              float* __restrict__ segsum) {
  int e = blockIdx.x * blockDim.x + threadIdx.x;
  if (e >= E) return;
  int dr = dst[e] + dstOff;
  float ex = expf(score[e] - key_f32(segmax[dr]));
  score[e] = ex;
  atomicAdd(&segsum[dr], ex);
}

// pass C: out[dst] += scale * a * H[src]
__global__ void __launch_bounds__(256)
k_edge_aggregate(const int* __restrict__ src, const int* __restrict__ dst, int E,
                 int srcOff, int dstOff, float scale,
                 const float* __restrict__ H, const float* __restrict__ score,
                 const float* __restrict__ segsum, float* __restrict__ out) {
  int e = blockIdx.x * 8 + (threadIdx.x >> 5);
  if (e >= E) return;
  int lane = threadIdx.x & 31;
  int sr = src[e] + srcOff;
  int dr = dst[e] + dstOff;
  float a = scale * score[e] / segsum[dr];
  float4 h4 = ((const float4*)(H + (size_t)sr * DIM))[lane];
  float* orow = out + (size_t)dr * DIM + lane * 4;
  atomicAdd(orow + 0, h4.x * a);
  atomicAdd(orow + 1, h4.y * a);
  atomicAdd(orow + 2, h4.z * a);
  atomicAdd(orow + 3, h4.w * a);
}

// ------------------------------------------------------------------ prediction
__global__ void __launch_bounds__(256)
k_predict(const float* __restrict__ h, const float* __restrict__ Wp,
          const float* __restrict__ bp, float* __restrict__ out, int N) {
  int n = blockIdx.x * 8 + (threadIdx.x >> 5);
  if (n >= N) return;
  int lane = threadIdx.x & 31;
  float4 h4 = ((const float4*)(h + (size_t)n * DIM))[lane];
  float4 w4 = ((const float4*)Wp)[lane];
  float s = h4.x * w4.x + h4.y * w4.y + h4.z * w4.z + h4.w * w4.w;
#pragma unroll
  for (int o = 16; o > 0; o >>= 1) s += __shfl_xor(s, o, 32);
  if (lane == 0) {
    float x = s + bp[0];
    out[n] = (x > 0.f) ? x + log1pf(expf(-x)) : log1pf(expf(x));
  }
}

// ================================================================== host
extern "C" void kernel_launch(void* const* d_in, const int* in_sizes, int n_in,
                              void* d_out, int out_size, void* d_ws, size_t ws_size,
                              hipStream_t stream) {
  (void)in_sizes; (void)n_in; (void)out_size; (void)ws_size;
  // pytree flatten order (jax sorts dict keys): W_pred, b_pred, layers[0..1];
  // per layer, etypes ap_ap < ap_sta < sta_ap; per etype:
  //   W_fij, W_ni, W_nj, W_node, attn, b_node, bias_e
  const float* Wpred = (const float*)d_in[0];
  const float* bpred = (const float*)d_in[1];
  auto P = [&](int l, int t, int j) -> const float* {
    return (const float*)d_in[2 + l * 21 + t * 7 + j];
  };
  const float* feat_ap  = (const float*)d_in[44];
  const float* feat_sta = (const float*)d_in[45];
  const float* efeat[3] = {(const float*)d_in[46], (const float*)d_in[47],
                           (const float*)d_in[48]};
  const int* src[3] = {(const int*)d_in[49], (const int*)d_in[51], (const int*)d_in[53]};
  const int* dst[3] = {(const int*)d_in[50], (const int*)d_in[52], (const int*)d_in[54]};
  const int Ecnt[3] = {E_AA, E_AS, E_SA};
  const long long Eoff[3] = {0, E_AA, (long long)E_AA + E_AS};
  // unified hn-order node space (AP rows first); 0.5*(h_aa+h_sa) folded in scale
  const int   srcOff[3] = {0, 0, N_AP};
  const int   dstOff[3] = {0, N_AP, 0};
  const float scl[3]    = {0.5f, 1.0f, 0.5f};

  char* wsb = (char*)d_ws;
  size_t off = 0;
  auto alloc = [&](size_t bytes) -> char* {
    char* p = wsb + off;
    off = (off + bytes + 255) & ~(size_t)255;
    return p;
  };
  float*    node_in   = (float*)alloc((size_t)N_NODE * DIM * 4);
  float*    node_next = (float*)alloc((size_t)N_NODE * DIM * 4);
  float*    Fb        = (float*)alloc((size_t)N_NODE * DIM * 4);
  float*    Gb        = (float*)alloc((size_t)N_NODE * DIM * 4);
  float*    Hb        = (float*)alloc((size_t)N_NODE * DIM * 4);
  float*    eA        = (float*)alloc((size_t)E_TOT * DIM * 4); // BN'ed edge feats
  float*    eB        = (float*)alloc((size_t)E_TOT * DIM * 4); // f_out / edge proj
  float*    esc       = (float*)alloc((size_t)E_TOT * 4);
  unsigned* segmax    = (unsigned*)alloc((size_t)N_NODE * 4);
  float*    segsum    = (float*)alloc((size_t)N_NODE * 4);
  float*    stats     = (float*)alloc(2 * DIM * 4);
  float*    Wt        = (float*)alloc((size_t)DIM * DIM * 4); // transposed weights

  auto fillf = [&](float* p, float v, long long n) {
    k_fill_f32<<<(unsigned)((n + 255) / 256), 256, 0, stream>>>(p, v, n);
  };
  auto fillu = [&](unsigned* p, unsigned v, long long n) {
    k_fill_u32<<<(unsigned)((n + 255) / 256), 256, 0, stream>>>(p, v, n);
  };
  auto bnstats = [&](const float* x, int N, int D) {
    k_bn_stats<<<2048, 128, 0, stream>>>(x, N, D, stats);
  };
  auto bnapply = [&](const float* x, long long rows, int D, int Dout, float ninv,
                     float* y) {
    long long total = rows * D;
    k_bn_apply<<<(unsigned)((total + 255) / 256), 256, 0, stream>>>(
        x, total, D, Dout, stats, ninv, y);
  };
  auto gemm = [&](const float* X, const float* W, const float* bias, float* Y,
                  int M, int K) {
    k_transpose<<<(K * DIM + 255) / 256, 256, 0, stream>>>(W, Wt, K);
    int blocks = ((M + 31) / 32 + 7) / 8;   // 32 rows/wave, 8 waves/block
    if (K == D_EDGE_IN)
      k_gemm_wmma<D_EDGE_IN><<<blocks, 256, 0, stream>>>(X, Wt, bias, Y, M);
    else if (K == K_NODE_L0)
      k_gemm_wmma<K_NODE_L0><<<blocks, 256, 0, stream>>>(X, Wt, bias, Y, M);
    else
      k_gemm_wmma<DIM><<<blocks, 256, 0, stream>>>(X, Wt, bias, Y, M);
  };

  for (int l = 0; l < 2; ++l) {
    int Dn = (l == 0) ? D_NODE_IN : DIM;   // BN width (true feature count)
    int Kn = (l == 0) ? K_NODE_L0 : DIM;   // GEMM K (padded row stride)
    int Ke = (l == 0) ? D_EDGE_IN : DIM;

    // ---- node BN over concat(h_ap, h_sta) -> node_in (row stride Kn)
    fillf(stats, 0.f, 2 * DIM);
    if (l == 0) {
      fillf(node_in, 0.f, (long long)N_NODE * K_NODE_L0);   // zero the pad cols
      bnstats(feat_ap, N_AP, Dn);
      bnstats(feat_sta, N_STA, Dn);
      bnapply(feat_ap, N_AP, Dn, Kn, 1.f / N_NODE, node_in);
      bnapply(feat_sta, N_STA, Dn, Kn, 1.f / N_NODE, node_in + (size_t)N_AP * Kn);
    } else {
      bnstats(node_next, N_NODE, Dn);
      bnapply(node_next, N_NODE, Dn, Kn, 1.f / N_NODE, node_in);
    }

    // ---- edge BN over concat(e_aa, e_as, e_sa) -> eA (row stride Ke)
    fillf(stats, 0.f, 2 * DIM);
    if (l == 0) {
      for (int t = 0; t < 3; ++t) bnstats(efeat[t], Ecnt[t], Ke);
      for (int t = 0; t < 3; ++t)
        bnapply(efeat[t], Ecnt[t], Ke, Ke, 1.f / E_TOT, eA + Eoff[t] * Ke);
    } else {
      bnstats(eB, E_TOT, Ke);
      bnapply(eB, E_TOT, Ke, Ke, 1.f / E_TOT, eA);
    }

    // ---- accumulate new node features
    fillf(node_next, 0.f, (long long)N_NODE * DIM);
    for (int t = 0; t < 3; ++t) {
      int Mn = (t == 0) ? N_AP : N_NODE;   // ap_ap sees AP rows only (contiguous)
      gemm(node_in, P(l, t, 1), nullptr,    Fb, Mn, Kn);              // W_ni
      gemm(node_in, P(l, t, 2), nullptr,    Gb, Mn, Kn);              // W_nj
      gemm(node_in, P(l, t, 3), P(l, t, 5), Hb, Mn, Kn);              // W_node+b
      gemm(eA + Eoff[t] * Ke, P(l, t, 0), P(l, t, 6),                 // W_fij+bias_e
           eB + Eoff[t] * DIM, Ecnt[t], Ke);

      fillu(segmax, 0u, N_NODE);   // key-space floor: below every real key
      fillf(segsum, 0.f, N_NODE);
      int gE = (Ecnt[t] + 7) / 8;
      k_edge_attn<<<gE, 256, 0, stream>>>(
          src[t], dst[t], Ecnt[t], srcOff[t], dstOff[t], Fb, Gb,
          eB + Eoff[t] * DIM, P(l, t, 4), esc + Eoff[t], segmax);
      k_edge_expsum<<<(Ecnt[t] + 255) / 256, 256, 0, stream>>>(
          dst[t], Ecnt[t], dstOff[t], esc + Eoff[t], segmax, segsum);
      k_edge_aggregate<<<gE, 256, 0, stream>>>(
          src[t], dst[t], Ecnt[t], srcOff[t], dstOff[t], scl[t], Hb,
          esc + Eoff[t], segsum, node_next);
    }
  }

  k_predict<<<(N_NODE + 7) / 8, 256, 0, stream>>>(
      node_next, Wpred, bpred, (float*)d_out, N_NODE);
}